// TrajectoryDecoder_1211180778406
// MI455X (gfx1250) — compile-verified
//
#include <hip/hip_runtime.h>
#include <math.h>

typedef __attribute__((ext_vector_type(2))) float v2f;
typedef __attribute__((ext_vector_type(8))) float v8f;

#define NB   8192
#define NK   6
#define NT   30
#define HID  256
#define GH   256
#define BK   (NB*NK)
#define TM   64          // rows per workgroup
#define HS   260         // LDS h row stride (floats), bank-conflict-free
#define SS   292         // staging row stride (floats)
#define KIN  290         // 256 + 2 + 32
#define KCH  64          // 256/4 k-chunks for W_hh
#define KC0  73          // 292/4 k-chunks for W_h0 (padded)

// LDS layout (floats)
#define HA_OFF   0
#define HB_OFF   (TM*HS)                 // 16640 ; also init_in staging (needs TM*SS=18688)
#define PRED_OFF (HB_OFF + TM*SS)        // 35328
#define PREV_OFF (PRED_OFF + TM*NT*2)    // 39168
#define WOUT_OFF (PREV_OFF + TM*2)       // 39296
#define PART_OFF (WOUT_OFF + 2*GH)       // 39808 : 64 rows * 4 partials * 2
#define SH_FLOATS (PART_OFF + TM*4*2)    // 40320 floats = 161280 bytes

__device__ __forceinline__ float sigm(float x){ return 1.0f/(1.0f + __expf(-x)); }

// Pack W_hh (768x256, row-major) into per-lane WMMA-B fragment order:
// for tile nt (16 cols), k-chunk kc (4 k), lane: lanes 0-15 hold (k0,k0+1) of
// column nt*16+lane, lanes 16-31 hold (k0+2,k0+3).  B[k][n] = W_hh[n][k].
__global__ void pack_whh(const float* __restrict__ W, float* __restrict__ P){
  int tid = blockIdx.x*256 + threadIdx.x;
  if (tid >= 48*KCH*32) return;
  int lane = tid & 31;
  int kc   = (tid >> 5) % KCH;
  int nt   = tid / (KCH*32);
  int row  = nt*16 + (lane & 15);
  int k0   = kc*4 + ((lane < 16) ? 0 : 2);
  P[2*tid]   = W[row*GH + k0];
  P[2*tid+1] = W[row*GH + k0 + 1];
}

// Same for W_h0 (256x290), K zero-padded to 292.
__global__ void pack_wh0(const float* __restrict__ W, float* __restrict__ P){
  int tid = blockIdx.x*256 + threadIdx.x;
  if (tid >= 16*KC0*32) return;
  int lane = tid & 31;
  int kc   = (tid >> 5) % KC0;
  int nt   = tid / (KC0*32);
  int row  = nt*16 + (lane & 15);
  int k0   = kc*4 + ((lane < 16) ? 0 : 2);
  P[2*tid]   = (k0   < KIN) ? W[row*KIN + k0]     : 0.0f;
  P[2*tid+1] = (k0+1 < KIN) ? W[row*KIN + k0 + 1] : 0.0f;
}

__launch_bounds__(256)
__global__ void traj_gru(const float* __restrict__ ctx,
                         const float* __restrict__ goals,
                         const float* __restrict__ emb,
                         const float* __restrict__ b_h0,
                         const float* __restrict__ W_ih,
                         const float* __restrict__ b_ih,
                         const float* __restrict__ b_hh,
                         const float* __restrict__ W_out,
                         const float* __restrict__ b_out,
                         const float* __restrict__ Phh,
                         const float* __restrict__ Ph0,
                         float* __restrict__ out)
{
  extern __shared__ float sm[];
  const int tid  = threadIdx.x;
  const int w    = tid >> 5;        // wave 0..7, owns h-columns [32w, 32w+32)
  const int lane = tid & 31;
  const int ln   = lane & 15;
  const int hi   = (lane < 16) ? 0 : 1;
  const int koff = hi ? 2 : 0;
  const int row0 = blockIdx.x * TM;

  // per-lane gate constants (columns c = 32w+16s+ln), loaded once
  float wr0[2],wr1[2],wz0[2],wz1[2],wn0[2],wn1[2];
  float bir[2],biz[2],binn[2],bhr[2],bhz[2],bhn[2],bh0v[2];
#pragma unroll
  for (int s=0;s<2;++s){
    const int c = w*32 + s*16 + ln;
    wr0[s]=W_ih[2*c];        wr1[s]=W_ih[2*c+1];
    wz0[s]=W_ih[2*(c+GH)];   wz1[s]=W_ih[2*(c+GH)+1];
    wn0[s]=W_ih[2*(c+2*GH)]; wn1[s]=W_ih[2*(c+2*GH)+1];
    bir[s]=b_ih[c]; biz[s]=b_ih[c+GH]; binn[s]=b_ih[c+2*GH];
    bhr[s]=b_hh[c]; bhz[s]=b_hh[c+GH]; bhn[s]=b_hh[c+2*GH];
    bh0v[s]=b_h0[c];
  }

  // init prev positions = 0, stage W_out into LDS
  for (int i=tid; i<TM*2; i+=256) sm[PREV_OFF+i]=0.0f;
  for (int i=tid; i<2*GH; i+=256) sm[WOUT_OFF+i]=W_out[i];

  // stage init_in = [context | goal | intent | 0,0] into LDS (stride SS)
  for (int i=tid; i<TM*SS; i+=256){
    const int r = i / SS, c = i % SS;
    const int rg = row0 + r;
    const int bb = rg / NK;
    float v = 0.0f;
    if (c < HID)        v = ctx[bb*HID + c];
    else if (c < HID+2) v = goals[rg*2 + (c-HID)];
    else if (c < KIN)   v = emb[bb*32 + (c-HID-2)];
    sm[HB_OFF + r*SS + c] = v;
  }
  __syncthreads();

  // ---------------- h0 = init_in @ W_h0^T + b_h0 via V_WMMA_F32_16X16X4_F32
  // NOTE: keep this loop rolled (#pragma unroll 1): full unrolling pushed the
  // kernel past 256 VGPRs (s_set_vgpr_msb) and would throttle occupancy for
  // the 30x heavier recurrence loop below.
  {
    const float* pb0 = Ph0 + ((size_t)(2*w+0)*KC0*32 + lane)*2;
    const float* pb1 = Ph0 + ((size_t)(2*w+1)*KC0*32 + lane)*2;
#pragma unroll 1
    for (int m=0;m<4;++m){
      v8f c0 = {0,0,0,0,0,0,0,0}, c1 = {0,0,0,0,0,0,0,0};
      const float* pa = &sm[HB_OFF + (m*16 + ln)*SS + koff];
#pragma unroll 1
      for (int kc=0;kc<KC0;++kc){
        v2f av = *(const v2f*)(pa  + kc*4);
        v2f b0 = *(const v2f*)(pb0 + kc*64);
        v2f b1 = *(const v2f*)(pb1 + kc*64);
        c0 = __builtin_amdgcn_wmma_f32_16x16x4_f32(false, av, false, b0, (short)0, c0, false, false);
        c1 = __builtin_amdgcn_wmma_f32_16x16x4_f32(false, av, false, b1, (short)0, c1, false, false);
      }
      const int rb = m*16 + hi*8;
#pragma unroll
      for (int v=0; v<8; ++v){
        sm[HA_OFF + (rb+v)*HS + (w*32 + ln)]      = c0[v] + bh0v[0];
        sm[HA_OFF + (rb+v)*HS + (w*32 + 16 + ln)] = c1[v] + bh0v[1];
      }
    }
  }
  __syncthreads();

  // B-fragment base pointers for the 6 owned column tiles of W_hh (r,z,n × 2)
  const float* pB[6];
#pragma unroll
  for (int g=0; g<3; ++g)
#pragma unroll
    for (int s=0; s<2; ++s)
      pB[g*2+s] = Phh + ((size_t)((g*16 + 2*w + s)*KCH)*32 + lane)*2;

  float* hc  = &sm[HA_OFF];   // h_{t-1}
  float* hnx = &sm[HB_OFF];   // h_t (staging buffer is dead now)
  const float bo0 = b_out[0], bo1 = b_out[1];
  const int  drow = tid >> 2;         // delta-stage row (0..63)
  const int  dq   = tid & 3;          // delta-stage column phase (0..3)

#pragma unroll 1
  for (int t=0; t<NT; ++t){
#pragma unroll 1
    for (int mp=0; mp<2; ++mp){          // 2 row-tile pairs of 32 rows
      v8f acc[2][6];
#pragma unroll
      for (int mi=0;mi<2;++mi)
#pragma unroll
        for (int ct=0;ct<6;++ct){ v8f z={0,0,0,0,0,0,0,0}; acc[mi][ct]=z; }
      const float* pa0 = &hc[(mp*32 + ln)*HS + koff];
      const float* pa1 = &hc[(mp*32 + 16 + ln)*HS + koff];
#pragma unroll 2
      for (int kc=0;kc<KCH;++kc){
        v2f a0 = *(const v2f*)(pa0 + kc*4);
        v2f a1 = *(const v2f*)(pa1 + kc*4);
#pragma unroll
        for (int ct=0;ct<6;++ct){
          v2f bv = *(const v2f*)(pB[ct] + kc*64);
          acc[0][ct] = __builtin_amdgcn_wmma_f32_16x16x4_f32(false, a0, false, bv, (short)0, acc[0][ct], false, false);
          acc[1][ct] = __builtin_amdgcn_wmma_f32_16x16x4_f32(false, a1, false, bv, (short)0, acc[1][ct], false, false);
        }
      }
      // fused GRU gates (gi is rank-2 -> pure FMAs from per-lane constants)
#pragma unroll
      for (int mi=0;mi<2;++mi){
        const int rb = mp*32 + mi*16 + hi*8;
#pragma unroll
        for (int s=0;s<2;++s){
          const int col = w*32 + s*16 + ln;
#pragma unroll
          for (int v=0; v<8; ++v){
            const int rl = rb + v;
            v2f xy = *(const v2f*)&sm[PREV_OFF + rl*2];
            const float x = xy.x, y = xy.y;
            const float rr = sigm (wr0[s]*x + wr1[s]*y + bir[s]  + acc[mi][s][v]   + bhr[s]);
            const float zz = sigm (wz0[s]*x + wz1[s]*y + biz[s]  + acc[mi][2+s][v] + bhz[s]);
            const float nn = tanhf(wn0[s]*x + wn1[s]*y + binn[s] + rr*(acc[mi][4+s][v] + bhn[s]));
            const float ho = hc[rl*HS + col];
            hnx[rl*HS + col] = (1.0f - zz)*nn + zz*ho;
          }
        }
      }
    }
    __syncthreads();

    // delta = h_new @ W_out^T, parallel over all 256 threads:
    // thread (drow, dq) sums columns j = dq + 4*i  (bank-conflict-free:
    // bank = (4*drow + dq + 4*i) mod 64, distinct across a wave)
    {
      float dx = 0.0f, dy = 0.0f;
      const float* hr = &hnx[drow*HS];
#pragma unroll 4
      for (int i=0;i<GH/4;++i){
        const int j = dq + 4*i;
        const float hv = hr[j];
        dx += hv*sm[WOUT_OFF+j];
        dy += hv*sm[WOUT_OFF+GH+j];
      }
      sm[PART_OFF + tid*2]   = dx;
      sm[PART_OFF + tid*2+1] = dy;
    }
    __syncthreads();
    if (tid < TM){
      float dx = bo0, dy = bo1;
#pragma unroll
      for (int q=0;q<4;++q){
        dx += sm[PART_OFF + (tid*4+q)*2];
        dy += sm[PART_OFF + (tid*4+q)*2+1];
      }
      const float px = sm[PREV_OFF + 2*tid]   + dx;
      const float py = sm[PREV_OFF + 2*tid+1] + dy;
      sm[PREV_OFF + 2*tid]   = px;
      sm[PREV_OFF + 2*tid+1] = py;
      sm[PRED_OFF + tid*(NT*2) + 2*t]   = px;
      sm[PRED_OFF + tid*(NT*2) + 2*t+1] = py;
    }
    __syncthreads();
    float* tsw = hc; hc = hnx; hnx = tsw;
  }

  // final goal correction: out = preds + (goal - preds[T-1]) * (t+1)/T
  for (int i=tid; i<TM*NT*2; i+=256){
    const int r    = i/(NT*2);
    const int rest = i%(NT*2);
    const int tt   = rest >> 1;
    const int c    = rest & 1;
    const int rg   = row0 + r;
    const float corr = goals[rg*2+c] - sm[PRED_OFF + r*(NT*2) + (NT-1)*2 + c];
    out[(size_t)row0*(NT*2) + i] = sm[PRED_OFF + i] + corr*((float)(tt+1)*(1.0f/NT));
  }
}

extern "C" void kernel_launch(void* const* d_in, const int* in_sizes, int n_in,
                              void* d_out, int out_size, void* d_ws, size_t ws_size,
                              hipStream_t stream) {
  const float* ctx   = (const float*)d_in[0];
  const float* goals = (const float*)d_in[1];
  const float* emb   = (const float*)d_in[2];
  const float* W_h0  = (const float*)d_in[3];
  const float* b_h0  = (const float*)d_in[4];
  const float* W_ih  = (const float*)d_in[5];
  const float* W_hh  = (const float*)d_in[6];
  const float* b_ih  = (const float*)d_in[7];
  const float* b_hh  = (const float*)d_in[8];
  const float* W_out = (const float*)d_in[9];
  const float* b_out = (const float*)d_in[10];

  float* Phh = (float*)d_ws;                    // 48*64*64  = 196608 floats
  float* Ph0 = Phh + (size_t)48*KCH*64;         // 16*73*64  =  74752 floats
  float* out = (float*)d_out;

  pack_whh<<<(48*KCH*32 + 255)/256, 256, 0, stream>>>(W_hh, Phh);
  pack_wh0<<<(16*KC0*32 + 255)/256, 256, 0, stream>>>(W_h0, Ph0);
  traj_gru<<<BK/TM, 256, SH_FLOATS*sizeof(float), stream>>>(
      ctx, goals, emb, b_h0, W_ih, b_ih, b_hh, W_out, b_out, Phh, Ph0, out);
}